// ProposalTarget_33122787787644
// MI455X (gfx1250) — compile-verified
//
#include <hip/hip_runtime.h>
#include <stdint.h>

#define NUM_FG      128
#define NUM_BG      384
#define BATCH_ROIS  512
#define ROW_LEN     654      // 5 roi + 1 label + 324 targets + 324 weights
#define FG_THRESH   0.5f
#define BG_HI       0.5f
#define BG_LO       0.1f
#define NBINS       4096
#define LIST_CAP    4096
#define MAX_GT      256

typedef unsigned int u32;
typedef unsigned long long u64;

// Monotone bin for score s: bin 0 reserved for invalid (-1), bins 1..4095 for [0,1].
__device__ __forceinline__ int score_bin(float s) {
    if (s < 0.0f) return 0;
    int b = (int)(s * 4094.0f);
    b = b > 4094 ? 4094 : b;
    return b + 1;
}

// 64-bit key: larger key == (larger value, then smaller index).
__device__ __forceinline__ u64 make_key(float s, int idx) {
    u32 b = __float_as_uint(s);
    u32 u = (b & 0x80000000u) ? ~b : (b | 0x80000000u);
    return ((u64)u << 32) | (u32)(~(u32)idx);
}

// ---------------------------------------------------------------- K0: zero out + hist/params
__global__ void init_kernel(float* __restrict__ out, int out_n,
                            u32* __restrict__ zr, int zn) {
    int i = blockIdx.x * blockDim.x + threadIdx.x;
    int stride = gridDim.x * blockDim.x;
    for (int j = i; j < out_n; j += stride) out[j] = 0.0f;
    for (int j = i; j < zn;    j += stride) zr[j]  = 0u;
}

// ---------------------------------------------------------------- K1: IoU + argmax + histograms
__global__ void iou_argmax_kernel(const float* __restrict__ rois,
                                  const float* __restrict__ gt,
                                  int N, int NG,
                                  float* __restrict__ max_ov,
                                  int*   __restrict__ gt_assign,
                                  u32*   __restrict__ fg_hist,
                                  u32*   __restrict__ bg_hist) {
    // single LDS block so the raw gt table sits at LDS offset 0 (async-copy target)
    __shared__ float smem[MAX_GT * 5 + MAX_GT];
    float* sgt   = smem;                 // NG x 5 raw gt_boxes
    float* sarea = smem + MAX_GT * 5;    // per-gt area

    const int tid = threadIdx.x;

    // --- CDNA5 async global->LDS staging of the GT table (tracked by ASYNCcnt) ---
    {
        int nbytes = NG * 5 * 4;         // NG=256 -> 5120 B
        int nchunk = nbytes >> 4;        // 16-byte chunks
        for (int c = tid; c < nchunk; c += blockDim.x) {
            u32 lds_off = (u32)(c * 16); // sgt is at LDS offset 0
            u32 g_off   = (u32)(c * 16);
            asm volatile("global_load_async_to_lds_b128 %0, %1, %2"
                         :: "v"(lds_off), "v"(g_off), "s"(gt)
                         : "memory");
        }
        asm volatile("s_wait_asynccnt 0x0" ::: "memory");
    }
    __syncthreads();
    for (int g = tid; g < NG; g += blockDim.x) {
        float x1 = sgt[g*5+0], y1 = sgt[g*5+1], x2 = sgt[g*5+2], y2 = sgt[g*5+3];
        sarea[g] = (x2 - x1 + 1.0f) * (y2 - y1 + 1.0f);
    }
    __syncthreads();

    for (int i = blockIdx.x * blockDim.x + tid; i < N; i += gridDim.x * blockDim.x) {
        float bx1 = rois[(size_t)i*5+1];
        float by1 = rois[(size_t)i*5+2];
        float bx2 = rois[(size_t)i*5+3];
        float by2 = rois[(size_t)i*5+4];
        float areab = (bx2 - bx1 + 1.0f) * (by2 - by1 + 1.0f);
        float best = -1e30f; int bestg = 0;
        for (int g = 0; g < NG; ++g) {   // uniform g -> LDS broadcast, no bank conflicts
            float gx1 = sgt[g*5+0], gy1 = sgt[g*5+1], gx2 = sgt[g*5+2], gy2 = sgt[g*5+3];
            float iw = fminf(bx2, gx2) - fmaxf(bx1, gx1) + 1.0f;
            float ih = fminf(by2, gy2) - fmaxf(by1, gy1) + 1.0f;
            iw = fmaxf(iw, 0.0f); ih = fmaxf(ih, 0.0f);
            float inter = iw * ih;
            float iou = inter / (areab + sarea[g] - inter);
            if (iou > best) { best = iou; bestg = g; }   // strict > : first-max wins (argmax)
        }
        max_ov[i]    = best;
        gt_assign[i] = bestg;
        float fs = (best >= FG_THRESH) ? best : -1.0f;
        float bs = (best < BG_HI && best >= BG_LO) ? best : -1.0f;
        if (fs >= 0.0f) atomicAdd(&fg_hist[score_bin(fs)], 1u);
        if (bs >= 0.0f) atomicAdd(&bg_hist[score_bin(bs)], 1u);
    }
}

// ---------------------------------------------------------------- K2: find K-th-element bin
__global__ void threshold_kernel(const u32* __restrict__ fg_hist,
                                 const u32* __restrict__ bg_hist,
                                 u32* __restrict__ params) {
    int tid = threadIdx.x;
    if (tid > 1) return;
    const u32* h = (tid == 0) ? fg_hist : bg_hist;
    u32 K = (tid == 0) ? NUM_FG : NUM_BG;
    u32 c = 0, t = 1; bool found = false;
    for (int b = NBINS - 1; b >= 1; --b) {
        c += h[b];
        if (!found && c >= K) { t = (u32)b; found = true; }
    }
    params[tid]     = t;   // threshold bin (>=1)
    params[2 + tid] = c;   // total valid candidates
}

// ---------------------------------------------------------------- K3: compact candidates >= threshold
__global__ void collect_kernel(const float* __restrict__ max_ov, int N,
                               const u32* __restrict__ params,
                               u32* __restrict__ cnts,        // params+4
                               u64* __restrict__ fg_list,
                               u64* __restrict__ bg_list) {
    u32 t_fg = params[0], t_bg = params[1];
    for (int i = blockIdx.x * blockDim.x + threadIdx.x; i < N;
         i += gridDim.x * blockDim.x) {
        float m  = max_ov[i];
        float fs = (m >= FG_THRESH) ? m : -1.0f;
        float bs = (m < BG_HI && m >= BG_LO) ? m : -1.0f;
        if (fs >= 0.0f && (u32)score_bin(fs) >= t_fg) {
            u32 p = atomicAdd(&cnts[0], 1u);
            if (p < LIST_CAP) fg_list[p] = make_key(fs, i);
        }
        if (bs >= 0.0f && (u32)score_bin(bs) >= t_bg) {
            u32 p = atomicAdd(&cnts[1], 1u);
            if (p < LIST_CAP) bg_list[p] = make_key(bs, i);
        }
    }
}

// ---------------------------------------------------------------- K4: LDS bitonic sort -> keep[]
__global__ void sort_select_kernel(const u64* __restrict__ fg_list,
                                   const u64* __restrict__ bg_list,
                                   const u32* __restrict__ params,
                                   const float* __restrict__ max_ov, int N,
                                   int* __restrict__ keep) {
    __shared__ u64 sk[LIST_CAP];                  // 32 KB of 320 KB WGP LDS
    const bool isFg = (blockIdx.x == 0);
    const u64* list = isFg ? fg_list : bg_list;
    const int  K    = isFg ? NUM_FG : NUM_BG;
    const int  base = isFg ? 0 : NUM_FG;
    u32 cnt = params[isFg ? 4 : 5];
    int M = (int)(cnt < (u32)LIST_CAP ? cnt : (u32)LIST_CAP);
    const int tid = threadIdx.x;

    for (int i = tid; i < LIST_CAP; i += blockDim.x)
        sk[i] = (i < M) ? list[i] : 0ull;         // real keys have MSB set -> pad sorts last
    __syncthreads();

    for (int k = 2; k <= LIST_CAP; k <<= 1) {
        for (int j = k >> 1; j > 0; j >>= 1) {
            for (int idx = tid; idx < LIST_CAP; idx += blockDim.x) {
                int ixj = idx ^ j;
                if (ixj > idx) {
                    bool up = ((idx & k) == 0);
                    u64 a = sk[idx], b = sk[ixj];
                    if (up ? (a < b) : (a > b)) { sk[idx] = b; sk[ixj] = a; }  // descending
                }
            }
            __syncthreads();
        }
    }

    if (tid < K && tid < M)
        keep[base + tid] = (int)(~(u32)(sk[tid] & 0xffffffffull));
    __syncthreads();

    // Rare path: fewer than K valid candidates -> fill with smallest-index invalid ROIs
    // (matches jax top_k stable tie-break on the -1 scores). Single wave32 ordered scan.
    if (M < K && tid < 32) {
        int need = K - M, filled = 0;
        for (int b0 = 0; b0 < N && filled < need; b0 += 32) {
            int i = b0 + tid;
            bool inval = false;
            if (i < N) {
                float m = max_ov[i];
                float s = isFg ? (m >= FG_THRESH ? m : -1.0f)
                               : ((m < BG_HI && m >= BG_LO) ? m : -1.0f);
                inval = (s < 0.0f);
            }
            u64 mask = __ballot(inval) & 0xffffffffull;
            int rank = __popcll(mask & ((1ull << tid) - 1ull));
            if (inval && (filled + rank) < need)
                keep[base + M + filled + rank] = i;
            filled += __popcll(mask);
        }
    }
}

// ---------------------------------------------------------------- K5: gather + bbox transform + scatter
__global__ void assemble_kernel(const float* __restrict__ rois,
                                const float* __restrict__ gt,
                                const float* __restrict__ max_ov,
                                const int*   __restrict__ gt_assign,
                                const int*   __restrict__ keep,
                                float* __restrict__ out) {
    int r = blockIdx.x * blockDim.x + threadIdx.x;
    if (r >= BATCH_ROIS) return;
    int idx = keep[r];
    const float* rr = rois + (size_t)idx * 5;
    float x1 = rr[1], y1 = rr[2], x2 = rr[3], y2 = rr[4];
    size_t o = (size_t)r * ROW_LEN;
    out[o+0] = rr[0]; out[o+1] = x1; out[o+2] = y1; out[o+3] = x2; out[o+4] = y2;

    int g = gt_assign[idx];
    float lab = 0.0f;
    if (r < NUM_FG && max_ov[idx] >= FG_THRESH) lab = gt[(size_t)g*5+4];
    out[o+5] = lab;

    if (lab > 0.0f) {
        float ew  = fmaxf(x2 - x1 + 1.0f, 1.0f);
        float eh  = fmaxf(y2 - y1 + 1.0f, 1.0f);
        float ecx = x1 + 0.5f * ew, ecy = y1 + 0.5f * eh;
        float gx1 = gt[(size_t)g*5+0], gy1 = gt[(size_t)g*5+1];
        float gx2 = gt[(size_t)g*5+2], gy2 = gt[(size_t)g*5+3];
        float gw  = fmaxf(gx2 - gx1 + 1.0f, 1.0f);
        float gh  = fmaxf(gy2 - gy1 + 1.0f, 1.0f);
        float gcx = gx1 + 0.5f * gw, gcy = gy1 + 0.5f * gh;
        float t0 = (gcx - ecx) / ew, t1 = (gcy - ecy) / eh;
        float t2 = logf(gw / ew),    t3 = logf(gh / eh);
        int c = (int)lab;
        size_t bt = o + 6 + (size_t)c * 4;
        out[bt+0] = t0; out[bt+1] = t1; out[bt+2] = t2; out[bt+3] = t3;
        size_t bw = o + 6 + 324 + (size_t)c * 4;
        out[bw+0] = 1.0f; out[bw+1] = 1.0f; out[bw+2] = 1.0f; out[bw+3] = 1.0f;
    }
}

// ---------------------------------------------------------------- host launcher
extern "C" void kernel_launch(void* const* d_in, const int* in_sizes, int n_in,
                              void* d_out, int out_size, void* d_ws, size_t ws_size,
                              hipStream_t stream) {
    const float* rois = (const float*)d_in[0];
    const float* gt   = (const float*)d_in[1];
    float* out = (float*)d_out;
    int N  = in_sizes[0] / 5;
    int NG = in_sizes[1] / 5; if (NG > MAX_GT) NG = MAX_GT;

    char* w = (char*)d_ws;
    size_t off = 0;
    auto carve = [&](size_t bytes) -> void* {
        void* p = (void*)(w + off);
        off += bytes; off = (off + 15) & ~(size_t)15;
        return p;
    };
    float* max_ov  = (float*)carve((size_t)N * 4);
    int*   ga      = (int*)  carve((size_t)N * 4);
    u32*   fg_hist = (u32*)  carve((size_t)NBINS * 4);   // fg_hist,bg_hist,params contiguous
    u32*   bg_hist = (u32*)  carve((size_t)NBINS * 4);
    u32*   params  = (u32*)  carve(16 * 4);
    u64*   fg_list = (u64*)  carve((size_t)LIST_CAP * 8);
    u64*   bg_list = (u64*)  carve((size_t)LIST_CAP * 8);
    int*   keep    = (int*)  carve((size_t)BATCH_ROIS * 4);
    (void)ws_size; (void)n_in;

    int zn = NBINS * 2 + 16;                       // zero hist + params in one shot
    init_kernel<<<512, 256, 0, stream>>>(out, out_size, fg_hist, zn);

    int blocks = (N + 255) / 256;
    iou_argmax_kernel<<<blocks, 256, 0, stream>>>(rois, gt, N, NG, max_ov, ga,
                                                  fg_hist, bg_hist);
    threshold_kernel<<<1, 32, 0, stream>>>(fg_hist, bg_hist, params);
    collect_kernel<<<blocks, 256, 0, stream>>>(max_ov, N, params, params + 4,
                                               fg_list, bg_list);
    sort_select_kernel<<<2, 512, 0, stream>>>(fg_list, bg_list, params,
                                              max_ov, N, keep);
    assemble_kernel<<<(BATCH_ROIS + 255) / 256, 256, 0, stream>>>(rois, gt, max_ov,
                                                                  ga, keep, out);
}